// SimpleSSM_24962349924472
// MI455X (gfx1250) — compile-verified
//
#include <hip/hip_runtime.h>

// SSM scan: h_t = a*h_{t-1} + b*x_t ; y = c*h. BATCH=256, SEQ=65536, f32.
// 3 passes: (1) superchunk end-states (weighted reduction),
//           (2) per-row carry scan over superchunks,
//           (3) WMMA-tiled application H = X*L + a^{n+1}*carry.

#define BATCH 256
#define SEQ   65536
#define SCLEN 512                 // timesteps per superchunk
#define NSC   (SEQ / SCLEN)       // 128 superchunks per row
#define ROWT  16                  // batch rows per WMMA tile
#define LDS_STRIDE 132            // 128 + 4 floats pad (bank-conflict free)

typedef __attribute__((ext_vector_type(2))) float v2f;
typedef __attribute__((ext_vector_type(8))) float v8f;

__device__ __forceinline__ float powi(float b, int e) {
    float r = 1.0f;
    while (e > 0) { if (e & 1) r *= b; b *= b; e >>= 1; }
    return r;
}

// ---------------- Pass 1: E[r][k] = sum_{p<512} a^{511-p} * b * x[r][k*512+p]
__global__ void __launch_bounds__(256)
ssm_chunk_end(const float* __restrict__ x, const float* __restrict__ Ap,
              const float* __restrict__ Bp, float* __restrict__ E) {
    const float a = Ap[0], b = Bp[0];
    const int wave = blockIdx.x * (blockDim.x >> 5) + (threadIdx.x >> 5);
    const int lane = threadIdx.x & 31;
    const int r = wave >> 7;          // / NSC
    const int k = wave & (NSC - 1);

    const float4* p = (const float4*)(x + (size_t)r * SEQ + (size_t)k * SCLEN + lane * 16);
    float4 f0 = p[0], f1 = p[1], f2 = p[2], f3 = p[3];

    float a16 = a;                    // a^16 via 4 squarings
    #pragma unroll
    for (int i = 0; i < 4; ++i) a16 *= a16;
    float w = powi(a16, 31 - lane);   // global weight of this lane's last element
    float s = 0.0f;
    #define ACC(v) { s += (v) * w; w *= a; }
    ACC(f3.w) ACC(f3.z) ACC(f3.y) ACC(f3.x)
    ACC(f2.w) ACC(f2.z) ACC(f2.y) ACC(f2.x)
    ACC(f1.w) ACC(f1.z) ACC(f1.y) ACC(f1.x)
    ACC(f0.w) ACC(f0.z) ACC(f0.y) ACC(f0.x)
    #undef ACC
    s *= b;
    #pragma unroll
    for (int off = 16; off; off >>= 1) s += __shfl_down(s, off);
    if (lane == 0) E[r * NSC + k] = s;
}

// ---------------- Pass 2: exclusive carries Csup[r][k] (one wave per row)
__global__ void __launch_bounds__(256)
ssm_scan_carries(const float* __restrict__ E, const float* __restrict__ Ap,
                 float* __restrict__ Csup) {
    const int r    = blockIdx.x * (blockDim.x >> 5) + (threadIdx.x >> 5);
    const int lane = threadIdx.x & 31;
    const float a = Ap[0];
    float a512 = a;                   // a^512 via 9 squarings
    #pragma unroll
    for (int i = 0; i < 9; ++i) a512 *= a512;

    float F[4], Cloc[4];
    #pragma unroll
    for (int j = 0; j < 4; ++j) F[j] = E[r * NSC + lane * 4 + j];
    float t = 0.0f;
    #pragma unroll
    for (int j = 0; j < 4; ++j) { Cloc[j] = t; t = a512 * t + F[j]; }

    float A4 = a512 * a512; A4 *= A4; // decay of a 4-superchunk group
    float D = A4;
    #pragma unroll
    for (int off = 1; off < 32; off <<= 1) {   // inclusive linear-recurrence scan
        float u = __shfl_up(t, off);
        if (lane >= off) t += D * u;
        D *= D;
    }
    float ex = __shfl_up(t, 1);
    if (lane == 0) ex = 0.0f;
    float dj = 1.0f;
    #pragma unroll
    for (int j = 0; j < 4; ++j) {
        Csup[r * NSC + lane * 4 + j] = Cloc[j] + dj * ex;
        dj *= a512;
    }
}

// ---------------- Pass 3: WMMA tiles, one wave per (16-row tile, superchunk)
__global__ void __launch_bounds__(32)
ssm_apply(const float* __restrict__ x, const float* __restrict__ Ap,
          const float* __restrict__ Bp, const float* __restrict__ Cp,
          const float* __restrict__ Csup, float* __restrict__ outp,
          float* __restrict__ stp) {
    __shared__ __align__(16) float lds[ROWT * LDS_STRIDE];
    const float a = Ap[0], b = Bp[0], cc = Cp[0];
    const int lane = threadIdx.x;         // blockDim == 32
    const int half = lane >> 4;
    const int n    = lane & 15;           // N (time within tile) for B/C/D frags
    const int rt = blockIdx.x >> 7;       // / NSC
    const int k  = blockIdx.x & (NSC - 1);
    const int r0 = rt * ROWT;

    // Constant B frags: L[s][n] = a^{n-s} (n>=s). VGPR j holds K = j + 2*half.
    float Bf[4][2];
    #pragma unroll
    for (int q = 0; q < 4; ++q)
        #pragma unroll
        for (int j = 0; j < 2; ++j) {
            int s = 4 * q + j + 2 * half;
            Bf[q][j] = (n >= s) ? powi(a, n - s) : 0.0f;
        }
    const float an1 = powi(a, n + 1);     // a^{n+1} for the carry column

    // Carry per D-VGPR row m = v + 8*half, uniform across each half-wave.
    float carr[8];
    #pragma unroll
    for (int v = 0; v < 8; ++v)
        carr[v] = Csup[(size_t)(r0 + v + 8 * half) * NSC + k];

    for (int st = 0; st < SCLEN / 128; ++st) {
        const size_t tbase = (size_t)k * SCLEN + (size_t)st * 128;
        // Stage 16 rows x 128 timesteps: 512B-contiguous b128 loads per row.
        #pragma unroll 4
        for (int rr = 0; rr < ROWT; ++rr) {
            const float* src = x + (size_t)(r0 + rr) * SEQ + tbase + lane * 4;
            float4 xv = *(const float4*)src;
            if (st + 1 < SCLEN / 128) __builtin_prefetch(src + 128, 0, 0);
            xv.x *= b; xv.y *= b; xv.z *= b; xv.w *= b;
            *(float4*)&lds[rr * LDS_STRIDE + lane * 4] = xv;
        }
        __syncthreads();

        #pragma unroll
        for (int tt = 0; tt < 8; ++tt) {
            v8f d = {};
            #pragma unroll
            for (int v = 0; v < 8; ++v) d[v] = an1 * carr[v];   // C = a^{n+1}*carry
            #pragma unroll
            for (int q = 0; q < 4; ++q) {                       // K=16 in 4 steps
                const int tloc = tt * 16 + 4 * q + 2 * half;    // A: K = 2*half+{0,1}
                v2f av = *(const v2f*)&lds[n * LDS_STRIDE + tloc];
                v2f bv = { Bf[q][0], Bf[q][1] };
                d = __builtin_amdgcn_wmma_f32_16x16x4_f32(
                        false, av, false, bv, (short)0, d, false, false);
            }
            __syncthreads();  // all X reads of this tile region done before overwrite
            #pragma unroll
            for (int v = 0; v < 8; ++v)                         // H -> LDS (row-major)
                lds[(v + 8 * half) * LDS_STRIDE + tt * 16 + n] = d[v];
            const int srcLane = 15 | (lane & 16);               // H[:,15] -> next carry
            #pragma unroll
            for (int v = 0; v < 8; ++v)
                carr[v] = __shfl(d[v], srcLane);
        }
        __syncthreads();

        // Coalesced b128 stores of states and outputs.
        #pragma unroll 4
        for (int rr = 0; rr < ROWT; ++rr) {
            float4 hv = *(const float4*)&lds[rr * LDS_STRIDE + lane * 4];
            size_t idx = (size_t)(r0 + rr) * SEQ + tbase + lane * 4;
            *(float4*)&stp[idx] = hv;
            float4 ov; ov.x = cc * hv.x; ov.y = cc * hv.y;
            ov.z = cc * hv.z; ov.w = cc * hv.w;
            *(float4*)&outp[idx] = ov;
        }
        __syncthreads();
    }
}

extern "C" void kernel_launch(void* const* d_in, const int* in_sizes, int n_in,
                              void* d_out, int out_size, void* d_ws, size_t ws_size,
                              hipStream_t stream) {
    (void)in_sizes; (void)n_in; (void)out_size; (void)ws_size;
    const float* x  = (const float*)d_in[0];
    const float* A  = (const float*)d_in[1];
    const float* B  = (const float*)d_in[2];
    const float* C  = (const float*)d_in[3];
    float* outp = (float*)d_out;
    float* stp  = outp + (size_t)BATCH * SEQ;
    float* E    = (float*)d_ws;                 // BATCH*NSC f32 = 128 KB
    float* Csup = E + BATCH * NSC;              // BATCH*NSC f32 = 128 KB

    ssm_chunk_end   <<<BATCH * NSC / 8, 256, 0, stream>>>(x, A, B, E);
    ssm_scan_carries<<<BATCH / 8,       256, 0, stream>>>(E, A, Csup);
    ssm_apply       <<<(BATCH / ROWT) * NSC, 32, 0, stream>>>(x, A, B, C, Csup, outp, stp);
}